// S2Attention_43069932044788
// MI455X (gfx1250) — compile-verified
//
#include <hip/hip_runtime.h>
#include <hip/hip_bf16.h>
#include <math.h>

typedef float v2f __attribute__((ext_vector_type(2)));
typedef float v8f __attribute__((ext_vector_type(8)));

#define NSP 32768   // H*W*D
#define CC  192
#define C7  1344

// shift-variant id per (chunk k, channel group g): 0=identity, 1=h fwd,2=h bwd,
// 3=w fwd,4=w bwd,5=d fwd,6=d bwd.  axis = _ORDERS[k][g>>1], dir = g&1.
__device__ const int VT[6][6] = {
    {1, 2, 3, 4, 5, 6},
    {1, 2, 5, 6, 3, 4},
    {3, 4, 1, 2, 5, 6},
    {3, 4, 5, 6, 1, 2},
    {5, 6, 1, 2, 3, 4},
    {5, 6, 3, 4, 1, 2},
};

// ---------------- Kernel A: total + 6 face sums of x per (b, channel) --------
__global__ __launch_bounds__(256) void face_sums_kernel(
    const float* __restrict__ x, float* __restrict__ s8) {
  const int bj = blockIdx.x;                 // 0 .. 2*192-1
  const float* xp = x + (size_t)bj * NSP;
  float acc[7] = {0.f, 0.f, 0.f, 0.f, 0.f, 0.f, 0.f};
  for (int m = threadIdx.x; m < NSP; m += 256) {
    float v = xp[m];
    int d = m & 31, w = (m >> 5) & 31, h = m >> 10;
    acc[0] += v;
    if (h == 0)  acc[1] += v;
    if (h == 31) acc[2] += v;
    if (w == 0)  acc[3] += v;
    if (w == 31) acc[4] += v;
    if (d == 0)  acc[5] += v;
    if (d == 31) acc[6] += v;
  }
  __shared__ float red[256];
  for (int q = 0; q < 7; ++q) {
    red[threadIdx.x] = acc[q];
    __syncthreads();
    for (int s = 128; s > 0; s >>= 1) {
      if (threadIdx.x < s) red[threadIdx.x] += red[threadIdx.x + s];
      __syncthreads();
    }
    if (threadIdx.x == 0) s8[bj * 8 + q] = red[0];
    __syncthreads();
  }
}

// ---------------- Kernel B: a -> gelu-MLP -> softmax weights bar, beta -------
__global__ __launch_bounds__(192) void gate_kernel(
    const float* __restrict__ s8, const float* __restrict__ w1,
    const float* __restrict__ b1, const float* __restrict__ sw1,
    const float* __restrict__ sw2, float* __restrict__ barbuf,
    float* __restrict__ betabuf) {
  const int b = blockIdx.x;
  const int c = threadIdx.x;  // 0..191
  __shared__ float sv[7][CC];
  __shared__ float avec[CC];
  __shared__ float gvec[CC];

  {
    const float* s = s8 + (size_t)(b * CC + c) * 8;
    float S = s[0], fh0 = s[1], fhM = s[2], fw0 = s[3], fwM = s[4],
          fd0 = s[5], fdM = s[6];
    sv[0][c] = S;
    sv[1][c] = S + fh0 - fhM;  sv[2][c] = S - fh0 + fhM;
    sv[3][c] = S + fw0 - fwM;  sv[4][c] = S - fw0 + fwM;
    sv[5][c] = S + fd0 - fdM;  sv[6][c] = S - fd0 + fdM;
  }
  __syncthreads();

  // a[b,c] = sum_k ( s_v(k,c) . w1[:, k*C+c]  +  N * b1[k*C+c] )
  const int g = c >> 5;
  float a = 0.f;
  for (int k = 0; k < 7; ++k) {
    const int vid = (k == 6) ? 0 : VT[k][g];
    const float* svv = sv[vid];
    const float* wcol = w1 + k * CC + c;
    float t = 0.f;
    for (int j = 0; j < CC; ++j) t += svv[j] * wcol[(size_t)j * C7];
    a += t + 32768.0f * b1[k * CC + c];
  }
  avec[c] = a;
  __syncthreads();

  // gelu(a @ sa_w1)   (exact erf gelu)
  {
    float t = 0.f;
    for (int j = 0; j < CC; ++j) t += avec[j] * sw1[j * CC + c];
    gvec[c] = 0.5f * t * (1.0f + erff(t * 0.70710678118654752440f));
  }
  __syncthreads();

  // hat = g @ sa_w2 ; softmax over k ; beta = sum_k bar*b1
  float hat[7];
  for (int k = 0; k < 7; ++k) {
    float hv = 0.f;
    const float* wcol = sw2 + k * CC + c;
    for (int j = 0; j < CC; ++j) hv += gvec[j] * wcol[(size_t)j * C7];
    hat[k] = hv;
  }
  float mx = hat[0];
  for (int k = 1; k < 7; ++k) mx = fmaxf(mx, hat[k]);
  float se = 0.f;
  for (int k = 0; k < 7; ++k) { hat[k] = expf(hat[k] - mx); se += hat[k]; }
  const float inv = 1.f / se;
  float beta = 0.f;
  for (int k = 0; k < 7; ++k) {
    float bk = hat[k] * inv;
    barbuf[(b * 7 + k) * CC + c] = bk;
    beta += bk * b1[k * CC + c];
  }
  betabuf[b * CC + c] = beta;
}

// ---------------- Kernel C: fused gathered-GEMM1 * bar  +  GEMM2 -------------
#define ZSTRIDE 193          // z rows: 16 x 193 = 3088 dwords (bank-safe)
#define OSTRIDE 17           // out transpose: 192 x 17 = 3264 dwords
#define WAVE_LDS 3264        // dwords per wave (max of both uses)

static __device__ inline v8f v8zero() {
  v8f z = {0.f, 0.f, 0.f, 0.f, 0.f, 0.f, 0.f, 0.f};
  return z;
}

__global__ __launch_bounds__(128) void s2a_main_kernel(
    const float* __restrict__ x, const float* __restrict__ w1,
    const float* __restrict__ w2, const float* __restrict__ b2,
    const float* __restrict__ barbuf, const float* __restrict__ betabuf,
    float* __restrict__ out) {
  __shared__ float smem[4 * WAVE_LDS];
  const int lane  = threadIdx.x & 31;
  const int wib   = threadIdx.x >> 5;
  const int gwave = blockIdx.x * 4 + wib;
  const int b     = gwave >> 11;        // / 2048
  const int tile  = gwave & 2047;
  const int n0    = tile << 4;
  const int row   = lane & 15;
  const int half  = lane >> 4;
  float* zl = smem + wib * WAVE_LDS;

  // per-lane clamped-shift source indices for the 7 variants
  const int n  = n0 + row;
  const int dd = n & 31, ww = (n >> 5) & 31, hh = n >> 10;
  int s0 = n;
  int s1 = n - ((hh > 0)  ? 1024 : 0);
  int s2 = n + ((hh < 31) ? 1024 : 0);
  int s3 = n - ((ww > 0)  ? 32 : 0);
  int s4 = n + ((ww < 31) ? 32 : 0);
  int s5 = n - ((dd > 0)  ? 1 : 0);
  int s6 = n + ((dd < 31) ? 1 : 0);

  const float* xb = x + (size_t)b * CC * NSP;

  v8f zacc[12];
#pragma unroll
  for (int t = 0; t < 12; ++t) zacc[t] = v8zero();

  // ---- GEMM1 fused with shift-gather (A addressing) and bar scaling ----
  // k stays rolled (zacc indices independent of k); g is FULLY UNROLLED so all
  // zacc[] indices are compile-time constants -> accumulators stay in VGPRs.
#pragma unroll 1
  for (int k = 0; k < 7; ++k) {
    const int kb = k * CC;
#pragma unroll
    for (int g = 0; g < 6; ++g) {
      const int vid = (k == 6) ? 0 : VT[k][g];
      int srcn = s0;
      srcn = (vid == 1) ? s1 : srcn;
      srcn = (vid == 2) ? s2 : srcn;
      srcn = (vid == 3) ? s3 : srcn;
      srcn = (vid == 4) ? s4 : srcn;
      srcn = (vid == 5) ? s5 : srcn;
      srcn = (vid == 6) ? s6 : srcn;
      const float* xa = xb + srcn;
      const int col0 = kb + g * 32 + row;

      v8f acc0 = v8zero();
      v8f acc1 = v8zero();
#pragma unroll 4
      for (int j0 = 0; j0 < CC; j0 += 4) {
        const int jA = j0 + half * 2;   // K sub-index for both A and B operands
        v2f av;
        av.x = xa[(size_t)jA << 15];
        av.y = xa[(size_t)(jA + 1) << 15];
        v2f bv0, bv1;
        bv0.x = w1[(size_t)jA * C7 + col0];
        bv0.y = w1[(size_t)(jA + 1) * C7 + col0];
        bv1.x = w1[(size_t)jA * C7 + col0 + 16];
        bv1.y = w1[(size_t)(jA + 1) * C7 + col0 + 16];
        acc0 = __builtin_amdgcn_wmma_f32_16x16x4_f32(
            false, av, false, bv0, (short)0, acc0, false, false);
        acc1 = __builtin_amdgcn_wmma_f32_16x16x4_f32(
            false, av, false, bv1, (short)0, acc1, false, false);
      }
      const float bw0 = barbuf[(b * 7 + k) * CC + g * 32 + row];
      const float bw1 = barbuf[(b * 7 + k) * CC + g * 32 + 16 + row];
#pragma unroll
      for (int q = 0; q < 8; ++q) {
        zacc[g * 2][q]     += bw0 * acc0[q];
        zacc[g * 2 + 1][q] += bw1 * acc1[q];
      }
    }
  }

  // ---- add beta, spill z (C-layout) to LDS for A-layout re-read ----
#pragma unroll
  for (int t = 0; t < 12; ++t) {
    const float bv = betabuf[b * CC + t * 16 + row];
#pragma unroll
    for (int q = 0; q < 8; ++q) {
      zl[(q + 8 * half) * ZSTRIDE + t * 16 + row] = zacc[t][q] + bv;
    }
  }

  // ---- GEMM2: final = z @ w2 (A from LDS, same wave so DS order is safe) ----
  v8f oacc[12];
#pragma unroll
  for (int t = 0; t < 12; ++t) oacc[t] = v8zero();
#pragma unroll 1
  for (int j0 = 0; j0 < CC; j0 += 4) {
    const int jA = j0 + half * 2;
    v2f av;
    av.x = zl[row * ZSTRIDE + jA];
    av.y = zl[row * ZSTRIDE + jA + 1];
#pragma unroll
    for (int t = 0; t < 12; ++t) {
      v2f bv;
      bv.x = w2[(size_t)jA * CC + t * 16 + row];
      bv.y = w2[(size_t)(jA + 1) * CC + t * 16 + row];
      oacc[t] = __builtin_amdgcn_wmma_f32_16x16x4_f32(
          false, av, false, bv, (short)0, oacc[t], false, false);
    }
  }

  // ---- add b2, transpose through LDS, coalesced [b, c, n] store ----
#pragma unroll
  for (int t = 0; t < 12; ++t) {
    const float bb = b2[t * 16 + row];
    const int co = t * 16 + row;
#pragma unroll
    for (int q = 0; q < 8; ++q) {
      zl[co * OSTRIDE + (q + 8 * half)] = oacc[t][q] + bb;
    }
  }
  float* ob = out + (size_t)b * CC * NSP + n0;
#pragma unroll 1
  for (int idx = lane; idx < CC * 16; idx += 32) {
    const int co = idx >> 4;
    const int r  = idx & 15;
    ob[(size_t)co * NSP + r] = zl[co * OSTRIDE + r];
  }
}

// ---------------------------------------------------------------------------
extern "C" void kernel_launch(void* const* d_in, const int* in_sizes, int n_in,
                              void* d_out, int out_size, void* d_ws,
                              size_t ws_size, hipStream_t stream) {
  const float* x   = (const float*)d_in[0];
  const float* w1  = (const float*)d_in[1];
  const float* b1  = (const float*)d_in[2];
  const float* sw1 = (const float*)d_in[3];
  const float* sw2 = (const float*)d_in[4];
  const float* w2  = (const float*)d_in[5];
  const float* b2  = (const float*)d_in[6];
  float* out = (float*)d_out;

  float* ws      = (float*)d_ws;
  float* s8      = ws;          // 2*192*8  = 3072 floats
  float* barbuf  = ws + 3072;   // 2*7*192  = 2688 floats
  float* betabuf = ws + 5760;   // 2*192    = 384 floats   (total < 24.6 KB)

  face_sums_kernel<<<2 * CC, 256, 0, stream>>>(x, s8);
  gate_kernel<<<2, CC, 0, stream>>>(s8, w1, b1, sw1, sw2, barbuf, betabuf);
  // 2 batches * 2048 tiles of 16 rows, 4 waves per block
  s2a_main_kernel<<<(2 * 2048) / 4, 128, 0, stream>>>(x, w1, w2, b2, barbuf,
                                                      betabuf, out);
}